// AttentionAggregator_55267639165131
// MI455X (gfx1250) — compile-verified
//
#include <hip/hip_runtime.h>
#include <cstddef>

typedef float v2f __attribute__((ext_vector_type(2)));
typedef float v8f __attribute__((ext_vector_type(8)));

#define D_DIM 128
#define K_NEIGH 32
#define NODES_PER_WG 16
#define LEAKY 0.2f

// wa[i] = sum_o W[i][o] * a[o]   (feat_weights is [D_IN][D_OUT] row-major)
__global__ void wa_precompute(const float* __restrict__ W,
                              const float* __restrict__ a,
                              float* __restrict__ wa) {
  const int i = threadIdx.x;  // 0..127
  const float4* W4 = (const float4*)(W + (size_t)i * D_DIM);
  const float4* a4 = (const float4*)a;
  float acc = 0.f;
#pragma unroll
  for (int j = 0; j < D_DIM / 4; ++j) {
    float4 w = W4[j], av = a4[j];
    acc += w.x * av.x + w.y * av.y + w.z * av.z + w.w * av.w;
  }
  wa[i] = acc;
}

__global__ __launch_bounds__(256)
void gat_fused(const float* __restrict__ selfv,
               const float* __restrict__ neigh,
               const float* __restrict__ W,
               const float* __restrict__ attn,
               const float* __restrict__ bias,
               const float* __restrict__ wa_g,
               float* __restrict__ out,
               int nNodes) {
  __shared__ __align__(16) float wa_s[D_DIM];                 // 512 B
  __shared__ float part[8][K_NEIGH][K_NEIGH + 1];             // 33.8 KB, stride 33 (coprime w/ 64 banks)
  __shared__ __align__(16) float ylds[NODES_PER_WG][132];     // 8.4 KB, stride 132 -> conflict-free A-frag reads

  const int tid  = threadIdx.x;
  const int wave = tid >> 5;
  const int lane = tid & 31;

  // wa into LDS (from workspace, or recompute per-block fallback)
  if (wa_g != nullptr) {
    if (tid < D_DIM) wa_s[tid] = wa_g[tid];
  } else if (tid < D_DIM) {
    const float4* W4 = (const float4*)(W + (size_t)tid * D_DIM);
    const float4* a4 = (const float4*)attn;
    float acc = 0.f;
#pragma unroll
    for (int j = 0; j < D_DIM / 4; ++j) {
      float4 w = W4[j], av = a4[j];
      acc += w.x * av.x + w.y * av.y + w.z * av.z + w.w * av.w;
    }
    wa_s[tid] = acc;
  }
  __syncthreads();

  const float4 wa4 = ((const float4*)wa_s)[lane];  // lane's 4-col chunk of wa
  const int blockBase = blockIdx.x * NODES_PER_WG;

  // ---- streaming phase: each wave handles 2 nodes -> rows of ylds ----
  for (int rep = 0; rep < 2; ++rep) {
    const int local = wave * 2 + rep;
    int node = blockBase + local;
    if (node >= nNodes) node = nNodes - 1;  // pad rows compute garbage; stores guarded

    // self row + self logit (butterfly reduce across wave32)
    const float4 sv = ((const float4*)(selfv + (size_t)node * D_DIM))[lane];
    float ps = sv.x * wa4.x + sv.y * wa4.y + sv.z * wa4.z + sv.w * wa4.w;
#pragma unroll
    for (int m = 16; m >= 1; m >>= 1) ps += __shfl_xor(ps, m, 32);

    // 32 neighbor rows, one coalesced float4 per lane per row, kept in registers
    const float4* nb = (const float4*)(neigh + (size_t)node * K_NEIGH * D_DIM);
    float4 v[K_NEIGH];
#pragma unroll
    for (int k = 0; k < K_NEIGH; ++k) v[k] = nb[k * (D_DIM / 4) + lane];

    // per-lane partial dots -> LDS transpose-reduce (lane k sums row k)
#pragma unroll
    for (int k = 0; k < K_NEIGH; ++k) {
      part[wave][k][lane] =
          v[k].x * wa4.x + v[k].y * wa4.y + v[k].z * wa4.z + v[k].w * wa4.w;
    }
    float rowsum = 0.f;
#pragma unroll
    for (int j = 0; j < K_NEIGH; ++j) rowsum += part[wave][lane][j];

    // leaky-relu + numerically-stable softmax over the 32 neighbor logits
    float lg = rowsum + ps;
    lg = lg > 0.f ? lg : LEAKY * lg;
    float mx = lg;
#pragma unroll
    for (int m = 16; m >= 1; m >>= 1) mx = fmaxf(mx, __shfl_xor(mx, m, 32));
    const float e = __expf(lg - mx);
    float s = e;
#pragma unroll
    for (int m = 16; m >= 1; m >>= 1) s += __shfl_xor(s, m, 32);
    const float coef = e / s;  // lane k holds coef_k

    // ybar = self + sum_k coef_k * x_k   (all in registers)
    float4 acc = {0.f, 0.f, 0.f, 0.f};
#pragma unroll
    for (int k = 0; k < K_NEIGH; ++k) {
      const float ck = __shfl(coef, k, 32);
      acc.x += ck * v[k].x; acc.y += ck * v[k].y;
      acc.z += ck * v[k].z; acc.w += ck * v[k].w;
    }
    float4 y;
    y.x = sv.x + acc.x; y.y = sv.y + acc.y;
    y.z = sv.z + acc.z; y.w = sv.w + acc.w;
    *(float4*)&ylds[local][4 * lane] = y;   // (132*local + 4*lane)*4 is 16B aligned
  }
  __syncthreads();

  // ---- WMMA phase: out[16 x 128] = relu(ylds[16 x 128] * W[128 x 128] + bias)
  // wave w computes N-tile w (cols 16w..16w+15), K split into 32 steps of 4.
  const int h = lane >> 4;   // half-wave selector (K pair / M offset)
  const int c = lane & 15;   // A: M row;  B,C: N col within tile
  v8f C = {};
  const float* Wcol = W + 16 * wave + c;
#pragma unroll
  for (int kk = 0; kk < D_DIM / 4; ++kk) {
    // A 16x4 f32 frag: lane -> row c, K = 4kk + 2h + {0,1}  (conflict-free b64 LDS read)
    const v2f A = *(const v2f*)&ylds[c][4 * kk + 2 * h];
    // B 4x16 f32 frag: lane -> col c, K = 4kk + 2h + {0,1}  (L0/L2-hot, W is 64 KB)
    v2f B;
    B.x = Wcol[(size_t)(4 * kk + 2 * h) * D_DIM];
    B.y = Wcol[(size_t)(4 * kk + 2 * h + 1) * D_DIM];
    C = __builtin_amdgcn_wmma_f32_16x16x4_f32(false, A, false, B, (short)0, C,
                                              false, false);
  }
  const float bcol = bias[16 * wave + c];
#pragma unroll
  for (int i = 0; i < 8; ++i) {
    const int row = blockBase + i + 8 * h;   // C layout: M = i + 8*(lane/16), N = lane%16
    if (row < nNodes)
      out[(size_t)row * D_DIM + 16 * wave + c] = fmaxf(C[i] + bcol, 0.f);
  }
}

extern "C" void kernel_launch(void* const* d_in, const int* in_sizes, int n_in,
                              void* d_out, int out_size, void* d_ws, size_t ws_size,
                              hipStream_t stream) {
  const float* selfv = (const float*)d_in[0];   // [N, 128]
  const float* neigh = (const float*)d_in[1];   // [N*32, 128]
  const float* W     = (const float*)d_in[2];   // [128, 128]
  const float* attn  = (const float*)d_in[3];   // [128, 1]
  const float* bias  = (const float*)d_in[4];   // [128]
  float* out = (float*)d_out;                   // [N, 128]
  const int nNodes = in_sizes[0] / D_DIM;

  float* wa = nullptr;
  if (d_ws != nullptr && ws_size >= (size_t)(D_DIM * sizeof(float))) {
    wa = (float*)d_ws;
    wa_precompute<<<1, D_DIM, 0, stream>>>(W, attn, wa);
  }
  const int grid = (nNodes + NODES_PER_WG - 1) / NODES_PER_WG;
  gat_fused<<<grid, 256, 0, stream>>>(selfv, neigh, W, attn, bias, wa, out, nNodes);
}